// Graph_26714696581174
// MI455X (gfx1250) — compile-verified
//
#include <hip/hip_runtime.h>
#include <hip/hip_bf16.h>

typedef float v2f __attribute__((ext_vector_type(2)));
typedef float v8f __attribute__((ext_vector_type(8)));

#define B_   4
#define L_   4096
#define H_   256
#define NH_  8
#define T_   64
#define D_   32
#define WIN_ 16
#define WN_  33          // window entries (2*WIN+1)
#define SC_  97          // T_ + WN_
#define SCP_ 99          // padded score row (odd stride -> conflict-free)
#define MASK_FILL_ (-1.0e12f)

// ---------------------------------------------------------------------------
// Generic fp32 WMMA GEMM:  C = act( (A [+A2]) @ W + bias )
// Block: 128 threads (4 waves), 64x64 output tile, K staged in 32-chunks.
// Each wave owns a 16-row strip (4 C-tiles of 16x16), V_WMMA_F32_16X16X4_F32.
// ---------------------------------------------------------------------------
template<int ACT>
__global__ __launch_bounds__(128)
void gemm_wmma_f32(const float* __restrict__ A, const float* __restrict__ A2,
                   const float* __restrict__ W, const float* __restrict__ bias,
                   float* __restrict__ C, int M, int N, int K)
{
    __shared__ float As[64][33];
    __shared__ float Ws[32][64];

    const int tid  = threadIdx.x;
    const int w    = tid >> 5;
    const int lane = tid & 31;
    const int hf   = lane >> 4;   // half-wave selects K-pair / M-half
    const int p    = lane & 15;   // row (A) / col (B, C)
    const int r0   = blockIdx.x * 64;
    const int c0   = blockIdx.y * 64;

    v8f acc[4] = {};

    for (int kt = 0; kt < K; kt += 32) {
        for (int i = tid; i < 64 * 32; i += 128) {
            int row = i >> 5, col = i & 31;
            size_t gi = (size_t)(r0 + row) * K + kt + col;
            float v = A[gi];
            if (A2) v += A2[gi];
            As[row][col] = v;
        }
        for (int i = tid; i < 32 * 64; i += 128) {
            int row = i >> 6, col = i & 63;
            Ws[row][col] = W[(size_t)(kt + row) * N + c0 + col];
        }
        __syncthreads();

        for (int k = 0; k < 32; k += 4) {
            v2f a;
            a.x = As[16 * w + p][k + 2 * hf];
            a.y = As[16 * w + p][k + 2 * hf + 1];
            for (int nt = 0; nt < 4; ++nt) {
                v2f bb;
                bb.x = Ws[k + 2 * hf][nt * 16 + p];
                bb.y = Ws[k + 2 * hf + 1][nt * 16 + p];
                acc[nt] = __builtin_amdgcn_wmma_f32_16x16x4_f32(
                    false, a, false, bb, (short)0, acc[nt], false, false);
            }
        }
        __syncthreads();
    }

    for (int nt = 0; nt < 4; ++nt) {
        for (int r = 0; r < 8; ++r) {
            int m   = 8 * hf + r;
            int row = r0 + 16 * w + m;
            int col = c0 + nt * 16 + p;
            float v = acc[nt][r] + bias[col];
            if (ACT == 1) v = tanhf(v);
            C[(size_t)row * N + col] = v;
        }
    }
}

// ---------------------------------------------------------------------------
// Per-row head dots: upon_q[b,n,l], down_s[b,n,l], down_t[b,n,t]
// ---------------------------------------------------------------------------
__global__ void head_dots(const float* __restrict__ ctxh, const float* __restrict__ typesh,
                          const float* __restrict__ upon, const float* __restrict__ down,
                          float* __restrict__ uq, float* __restrict__ ds,
                          float* __restrict__ dt)
{
    int idx = blockIdx.x * blockDim.x + threadIdx.x;
    if (idx < B_ * NH_ * L_) {
        int l = idx % L_; int bn = idx / L_; int n = bn % NH_; int b = bn / NH_;
        const float* row = ctxh + (size_t)(b * L_ + l) * H_ + n * D_;
        float su = 0.f, sd = 0.f;
        for (int d = 0; d < D_; ++d) {
            su += row[d] * upon[n * D_ + d];
            sd += row[d] * down[n * D_ + d];
        }
        uq[idx] = su;
        ds[idx] = sd;
    }
    if (idx < B_ * NH_ * T_) {
        int t = idx % T_; int bn = idx / T_; int n = bn % NH_; int b = bn / NH_;
        const float* row = typesh + (size_t)(b * T_ + t) * H_ + n * D_;
        float sd = 0.f;
        for (int d = 0; d < D_; ++d) sd += row[d] * down[n * D_ + d];
        dt[idx] = sd;
    }
}

// ---------------------------------------------------------------------------
// Fused attention kernel.
// Block = 64 threads (2 waves); each wave handles 16 context rows of one head.
// blockIdx = (l_tile of 32 rows, head n, batch b).
// All matrix math via V_WMMA_F32_16X16X4_F32.
// ---------------------------------------------------------------------------
__global__ __launch_bounds__(64)
void attn_kernel(const float* __restrict__ ctxh, const float* __restrict__ typesh,
                 const float* __restrict__ cross, const unsigned char* __restrict__ cmask,
                 const float* __restrict__ uq_g, const float* __restrict__ ds_g,
                 const float* __restrict__ dt_g, float* __restrict__ update)
{
    __shared__ float ctx_s[48][D_ + 1];     // rows r0 .. r0+47 of context_h (head slice)
    __shared__ float typ_s[T_][D_ + 1];     // types_h head slice
    __shared__ float crs_s[D_][D_];         // cross[n]
    __shared__ float uq_s[64];              // rows r0-16 .. r0+47
    __shared__ float ds_s[64];
    __shared__ float dt_s[T_];
    __shared__ float q2_s[2][16][D_ + 1];   // per-wave q2 tile
    __shared__ float scr_s[2][16][SCP_];    // 97 scores -> weights per row
    __shared__ float psf_s[2][16][D_ + 1];  // raw band products, then banded A

    const int tid  = threadIdx.x;
    const int w    = tid >> 5;
    const int lane = tid & 31;
    const int hf   = lane >> 4;
    const int p    = lane & 15;
    const int n    = blockIdx.y;
    const int b    = blockIdx.z;
    const int r0   = blockIdx.x * 32;

    // ---- stage LDS ----
    for (int i = tid; i < D_ * D_; i += 64)
        crs_s[i >> 5][i & 31] = cross[(size_t)n * D_ * D_ + i];
    for (int i = tid; i < T_ * D_; i += 64) {
        int t = i >> 5, d = i & 31;
        typ_s[t][d] = typesh[(size_t)(b * T_ + t) * H_ + n * D_ + d];
    }
    for (int i = tid; i < T_; i += 64)
        dt_s[i] = dt_g[(size_t)(b * NH_ + n) * T_ + i];
    for (int i = tid; i < 48 * D_; i += 64) {
        int row = i >> 5, d = i & 31;
        int gr = r0 + row; if (gr >= L_) gr = L_ - 1;
        ctx_s[row][d] = ctxh[(size_t)(b * L_ + gr) * H_ + n * D_ + d];
    }
    for (int i = tid; i < 64; i += 64) {
        int gr = r0 - WIN_ + i;
        if (gr < 0) gr = 0; if (gr >= L_) gr = L_ - 1;
        uq_s[i] = uq_g[(size_t)(b * NH_ + n) * L_ + gr];
        ds_s[i] = ds_g[(size_t)(b * NH_ + n) * L_ + gr];
    }
    __syncthreads();

    const int rbase = 16 * w;   // ctx_s row of this wave's tile row m=0

    // ---- q2 = ctx_tile(16xD) @ cross(DxD) ----
    for (int nt = 0; nt < 2; ++nt) {
        v8f acc = {};
        for (int k = 0; k < D_; k += 4) {
            v2f a, bb;
            a.x  = ctx_s[rbase + p][k + 2 * hf];
            a.y  = ctx_s[rbase + p][k + 2 * hf + 1];
            bb.x = crs_s[k + 2 * hf][nt * 16 + p];
            bb.y = crs_s[k + 2 * hf + 1][nt * 16 + p];
            acc = __builtin_amdgcn_wmma_f32_16x16x4_f32(false, a, false, bb, (short)0, acc, false, false);
        }
        for (int r = 0; r < 8; ++r) q2_s[w][8 * hf + r][nt * 16 + p] = acc[r];
    }

    // ---- type scores: leaky(uq[l] + dt[t] + q2 @ types_h^T) ----
    for (int nt = 0; nt < 4; ++nt) {
        v8f acc = {};
        for (int k = 0; k < D_; k += 4) {
            v2f a, bb;
            a.x  = q2_s[w][p][k + 2 * hf];
            a.y  = q2_s[w][p][k + 2 * hf + 1];
            bb.x = typ_s[nt * 16 + p][k + 2 * hf];      // B[k][t] = types_h[t][k]
            bb.y = typ_s[nt * 16 + p][k + 2 * hf + 1];
            acc = __builtin_amdgcn_wmma_f32_16x16x4_f32(false, a, false, bb, (short)0, acc, false, false);
        }
        for (int r = 0; r < 8; ++r) {
            int m = 8 * hf + r, t = nt * 16 + p;
            float sc = uq_s[WIN_ + rbase + m] + dt_s[t] + acc[r];
            scr_s[w][m][t] = (sc >= 0.f) ? sc : 5.f * sc;
        }
    }

    // ---- raw band products P(16x32) = q2 @ ctx[r0+16w .. +31]^T ----
    for (int nt = 0; nt < 2; ++nt) {
        v8f acc = {};
        for (int k = 0; k < D_; k += 4) {
            v2f a, bb;
            a.x  = q2_s[w][p][k + 2 * hf];
            a.y  = q2_s[w][p][k + 2 * hf + 1];
            bb.x = ctx_s[rbase + nt * 16 + p][k + 2 * hf];   // B[k][c] = ctx[c][k]
            bb.y = ctx_s[rbase + nt * 16 + p][k + 2 * hf + 1];
            acc = __builtin_amdgcn_wmma_f32_16x16x4_f32(false, a, false, bb, (short)0, acc, false, false);
        }
        for (int r = 0; r < 8; ++r) psf_s[w][8 * hf + r][nt * 16 + p] = acc[r];
    }

    // ---- window scores + 97-way softmax + banded A (one lane per row) ----
    if (hf == 0) {
        int m = p;
        int l = r0 + 16 * w + m;
        for (int v = 0; v < WN_; ++v) {
            int ju = (v <= WIN_) ? 0 : (v - WIN_);   // off_u
            int jd = (v <  WIN_) ? (v - WIN_) : 0;   // off_d
            int iu = l + ju, id = l + jd;
            bool ok = (id >= 0) && (iu < L_);
            int iu_c = (iu < L_) ? iu : (iu - L_);   // idx_u % L
            ok = ok && (cmask[b * L_ + iu_c] != 0);
            float prod = psf_s[w][m][m + ju];
            float cc   = uq_s[WIN_ + rbase + m + ju] + ds_s[WIN_ + rbase + m + jd];
            float sc = cc + prod;
            sc = (sc >= 0.f) ? sc : 5.f * sc;
            scr_s[w][m][T_ + v] = ok ? sc : MASK_FILL_;
        }
        float mx = -3.4e38f;
        for (int i = 0; i < SC_; ++i) mx = fmaxf(mx, scr_s[w][m][i]);
        float sum = 0.f;
        for (int i = 0; i < SC_; ++i) {
            float e = expf(scr_s[w][m][i] - mx);
            scr_s[w][m][i] = e;
            sum += e;
        }
        float inv = 1.f / sum;
        for (int i = 0; i < SC_; ++i) scr_s[w][m][i] *= inv;
        // collapse window weights into banded A(16x32): A[m][c] = cw[c-m]
        float w0 = 0.f;
        for (int v = 0; v <= WIN_; ++v) w0 += scr_s[w][m][T_ + v];
        for (int c = 0; c < D_; ++c) {
            int j = c - m;
            float av = 0.f;
            if (j == 0)                    av = w0;
            else if (j >= 1 && j <= WIN_)  av = scr_s[w][m][T_ + WIN_ + j];
            psf_s[w][m][c] = av;
        }
    }

    // ---- update = Wt(16x64)@types(64x32) + A_band(16x32)@ctx(32x32) ----
    for (int nt = 0; nt < 2; ++nt) {
        v8f acc = {};
        for (int k = 0; k < T_; k += 4) {
            v2f a, bb;
            a.x  = scr_s[w][p][k + 2 * hf];
            a.y  = scr_s[w][p][k + 2 * hf + 1];
            bb.x = typ_s[k + 2 * hf][nt * 16 + p];
            bb.y = typ_s[k + 2 * hf + 1][nt * 16 + p];
            acc = __builtin_amdgcn_wmma_f32_16x16x4_f32(false, a, false, bb, (short)0, acc, false, false);
        }
        for (int k = 0; k < D_; k += 4) {
            v2f a, bb;
            a.x  = psf_s[w][p][k + 2 * hf];
            a.y  = psf_s[w][p][k + 2 * hf + 1];
            bb.x = ctx_s[rbase + k + 2 * hf][nt * 16 + p];
            bb.y = ctx_s[rbase + k + 2 * hf + 1][nt * 16 + p];
            acc = __builtin_amdgcn_wmma_f32_16x16x4_f32(false, a, false, bb, (short)0, acc, false, false);
        }
        for (int r = 0; r < 8; ++r) {
            int m = 8 * hf + r;
            int l = r0 + 16 * w + m;
            int d = nt * 16 + p;
            update[(size_t)(b * L_ + l) * H_ + n * D_ + d] = acc[r];
        }
    }
}

// ---------------------------------------------------------------------------
extern "C" void kernel_launch(void* const* d_in, const int* in_sizes, int n_in,
                              void* d_out, int out_size, void* d_ws, size_t ws_size,
                              hipStream_t stream)
{
    (void)in_sizes; (void)n_in; (void)out_size; (void)ws_size;

    const float*         context   = (const float*)d_in[0];          // (B,L,H)
    const float*         types     = (const float*)d_in[1];          // (B,T,H)
    const unsigned char* cmask     = (const unsigned char*)d_in[2];  // (B,L) bool
    const float*         W_types   = (const float*)d_in[3];
    const float*         b_types   = (const float*)d_in[4];
    const float*         W_context = (const float*)d_in[5];
    const float*         b_context = (const float*)d_in[6];
    const float*         upon      = (const float*)d_in[7];          // (NH,D)
    const float*         down      = (const float*)d_in[8];
    const float*         cross     = (const float*)d_in[9];          // (NH,D,D)
    const float*         W_out     = (const float*)d_in[10];
    const float*         b_out     = (const float*)d_in[11];
    float*               out       = (float*)d_out;                  // (B,L,H)

    float* ws      = (float*)d_ws;
    float* typesh  = ws;                           // B*T*H   = 65536
    float* ctxh    = typesh + (size_t)B_*T_*H_;    // B*L*H   = 4194304
    float* uq      = ctxh   + (size_t)B_*L_*H_;    // B*NH*L  = 131072
    float* dsv     = uq     + (size_t)B_*NH_*L_;
    float* dtv     = dsv    + (size_t)B_*NH_*L_;   // B*NH*T  = 2048
    float* upd     = dtv    + (size_t)B_*NH_*T_;   // B*L*H

    const int Mc = B_ * L_;   // 16384
    const int Mt = B_ * T_;   // 256

    // 1) types_h = types @ W_types + b_types
    gemm_wmma_f32<0><<<dim3(Mt / 64, H_ / 64), dim3(128), 0, stream>>>(
        types, nullptr, W_types, b_types, typesh, Mt, H_, H_);

    // 2) context_h = context @ W_context + b_context
    gemm_wmma_f32<0><<<dim3(Mc / 64, H_ / 64), dim3(128), 0, stream>>>(
        context, nullptr, W_context, b_context, ctxh, Mc, H_, H_);

    // 3) per-row dots (upon_q, down_s, down_t)
    head_dots<<<(B_ * NH_ * L_ + 255) / 256, dim3(256), 0, stream>>>(
        ctxh, typesh, upon, down, uq, dsv, dtv);

    // 4) fused attention -> update
    attn_kernel<<<dim3(L_ / 32, NH_, B_), dim3(64), 0, stream>>>(
        ctxh, typesh, cross, cmask, uq, dsv, dtv, upd);

    // 5) out = tanh((update + context) @ W_out + b_out)
    gemm_wmma_f32<1><<<dim3(Mc / 64, H_ / 64), dim3(128), 0, stream>>>(
        upd, context, W_out, b_out, out, Mc, H_, H_);
}